// PairwiseGenerativeRetrievalLoss_66984309948418
// MI455X (gfx1250) — compile-verified
//
#include <hip/hip_runtime.h>
#include <math.h>

// ---------------------------------------------------------------------------
// PairwiseGenerativeRetrievalLoss for MI455X (gfx1250).
// Bandwidth-bound row-scan: 2560 rows x 32001 f32 x 4 streams ~= 1.31 GB read,
// scalar output -> HBM-bound, ~56 us floor at 23.3 TB/s. Strategy: one 8-wave
// block per row, async global->LDS double-buffered tiles (ASYNCcnt pipeline),
// wave32 shuffle reductions, tiny finalize kernel for the cross-depth
// recursion. No GEMM structure -> WMMA not applicable; the CDNA5 path used is
// the async-to-LDS engine + split wait counters.
// ---------------------------------------------------------------------------

#define MAXD   5
#define BSZ    512
#define DTOK   32000
#define VTOK   32001
#define TILE   1024
#define BLK    256
#define NTILES ((VTOK + TILE - 1) / TILE)   // 32
#define NWAVES (BLK / 32)                   // 8
#define ALPHA_F 0.1f

struct RowOut {
  float A;     // sum e^{lpq} (1 - e^{ln})
  float B;     // sum e^{lpq} (1 - e^{ln}) * lpq
  float T;     // sum e^{lpq}
  float C;     // sum e^{ln} * ln
  float E;     // sum e^{ln} e^{lpq} * ln
  float Fs;    // sum_{v<D} (lp + ln + lq)
  float Es;    // lp[D] + ln[D] + lq[D]   (EOS column)
  float sum3;  // lp[nxt] + ln[nxt] + lq[nxt]
  float sm;    // exp(pair sum selected by sampling_target)
};

struct KLConsts { float c1[MAXD]; float a[MAXD]; float t3[MAXD]; };

#if defined(__has_builtin)
#  if __has_builtin(__builtin_amdgcn_global_load_async_to_lds_b32) && \
      __has_builtin(__builtin_amdgcn_s_wait_asynccnt)
#    define USE_ASYNC_LDS 1
#  endif
#  if __has_builtin(__builtin_amdgcn_s_wait_tensorcnt)
#    define HAVE_TENSORCNT 1
#  endif
#endif

#ifdef USE_ASYNC_LDS
__device__ __forceinline__ void async_b32(const float* g, float* l) {
  // This clang declares the builtin with int* parameters (per diagnostic);
  // CodeGen handles the address spaces.
  __builtin_amdgcn_global_load_async_to_lds_b32((int*)g, (int*)l, 0, 0);
}
#endif

__device__ __forceinline__ float wave_sum(float v) {
#pragma unroll
  for (int off = 16; off > 0; off >>= 1) v += __shfl_xor(v, off, 32);
  return v;
}

// Per-element accumulation body (shared by async and fallback paths).
// First-index tie-break on the Gumbel argmax matches jnp.argmax semantics.
#define ACC_BODY(vv, lp, lnv, lq, g)                                         \
  do {                                                                       \
    float lpq = (lp) + (lq);                                                 \
    float e1  = __expf(lpq);                                                 \
    float e2  = __expf(lnv);                                                 \
    float w   = e1 * (1.0f - e2);                                            \
    accA += w; accB += w * lpq; accT += e1;                                  \
    float ct = e2 * (lnv);                                                   \
    accC += ct; accE += ct * e1;                                             \
    if ((vv) < DTOK) {                                                       \
      accF += (lp) + (lnv) + (lq);                                           \
      float sel  = (st == 0) ? (lp) : ((st == 1) ? (lnv) : (lq));            \
      float cand = sel + (g);                                                \
      if (cand > bestV) { bestV = cand; bestI = (vv); }                      \
    } else {                                                                 \
      s_eos[0] = (lp); s_eos[1] = (lnv); s_eos[2] = (lq);                    \
    }                                                                        \
  } while (0)

__global__ __launch_bounds__(BLK)
void row_scan_kernel(const float* __restrict__ lp_all,
                     const float* __restrict__ ln_all,
                     const float* __restrict__ lq_all,
                     const float* __restrict__ g_all,
                     const int*   __restrict__ st_all,
                     RowOut* __restrict__ out) {
#ifdef HAVE_TENSORCNT
  __builtin_amdgcn_s_wait_tensorcnt(0);  // gfx1250 split-counter path (no-op here)
#endif
  const int row = blockIdx.x;            // row = depth*BSZ + b
  const int lid = threadIdx.x;
  const size_t offV = (size_t)row * VTOK;
  const size_t offD = (size_t)row * DTOK;
  const float* lpr = lp_all + offV;
  const float* lnr = ln_all + offV;
  const float* lqr = lq_all + offV;
  const float* gr  = g_all  + offD;
  const int st = st_all[row];

  float accA = 0.f, accB = 0.f, accT = 0.f, accC = 0.f, accE = 0.f, accF = 0.f;
  float bestV = -3.4e38f;
  int   bestI = 0x7fffffff;

  __shared__ float s_eos[3];
  __shared__ float s_part[NWAVES][7];
  __shared__ int   s_idx[NWAVES];

#ifdef USE_ASYNC_LDS
  __shared__ float sLp[2 * TILE], sLn[2 * TILE], sLq[2 * TILE], sG[2 * TILE];
  // Prime the pipeline: issue tile 0 into buffer 0.
  {
#pragma unroll
    for (int k = 0; k < TILE / BLK; ++k) {
      int i = lid + k * BLK;           // tile 0: v == i, always in range
      async_b32(lpr + i, sLp + i);
      async_b32(lnr + i, sLn + i);
      async_b32(lqr + i, sLq + i);
      async_b32(gr  + i, sG  + i);
    }
  }
  for (int t = 0; t < NTILES; ++t) {
    if (t + 1 < NTILES) {
      const int base = (t + 1) * TILE;
      const int bo   = ((t + 1) & 1) * TILE;
#pragma unroll
      for (int k = 0; k < TILE / BLK; ++k) {
        int i  = lid + k * BLK;
        int v  = base + i;
        int vc = (v < VTOK) ? v : (VTOK - 1);  // clamp: OOB-safe, masked later
        int gc = (v < DTOK) ? v : (DTOK - 1);
        async_b32(lpr + vc, sLp + bo + i);
        async_b32(lnr + vc, sLn + bo + i);
        async_b32(lqr + vc, sLq + bo + i);
        async_b32(gr  + gc, sG  + bo + i);
      }
      __builtin_amdgcn_s_wait_asynccnt(4 * (TILE / BLK));  // retire tile t
    } else {
      __builtin_amdgcn_s_wait_asynccnt(0);
    }
    __syncthreads();  // all waves' async copies of tile t visible in LDS
    const int bo = (t & 1) * TILE;
#pragma unroll
    for (int k = 0; k < TILE / BLK; ++k) {
      int i = lid + k * BLK;
      int v = t * TILE + i;
      float lp  = sLp[bo + i];
      float lnv = sLn[bo + i];
      float lq  = sLq[bo + i];
      float g   = sG[bo + i];
      if (v < VTOK) { ACC_BODY(v, lp, lnv, lq, g); }
    }
    __syncthreads();  // buffer reusable for tile t+2
  }
#else
  // Fallback: direct coalesced streaming (still bandwidth-optimal).
  for (int v = lid; v < VTOK; v += BLK) {
    float lp  = lpr[v];
    float lnv = lnr[v];
    float lq  = lqr[v];
    float g   = (v < DTOK) ? gr[v] : 0.f;
    ACC_BODY(v, lp, lnv, lq, g);
  }
#endif

  // Wave32 reductions.
  accA = wave_sum(accA); accB = wave_sum(accB); accT = wave_sum(accT);
  accC = wave_sum(accC); accE = wave_sum(accE); accF = wave_sum(accF);
#pragma unroll
  for (int off = 16; off > 0; off >>= 1) {
    float ov = __shfl_xor(bestV, off, 32);
    int   oi = __shfl_xor(bestI, off, 32);
    if (ov > bestV || (ov == bestV && oi < bestI)) { bestV = ov; bestI = oi; }
  }
  const int wid = lid >> 5, lane = lid & 31;
  if (lane == 0) {
    s_part[wid][0] = accA; s_part[wid][1] = accB; s_part[wid][2] = accT;
    s_part[wid][3] = accC; s_part[wid][4] = accE; s_part[wid][5] = accF;
    s_part[wid][6] = bestV; s_idx[wid] = bestI;
  }
  __syncthreads();
  if (lid == 0) {
    float A = 0.f, B = 0.f, T = 0.f, C = 0.f, E = 0.f, F = 0.f;
    float bv = -3.4e38f; int bi = 0x7fffffff;
    for (int w = 0; w < NWAVES; ++w) {
      A += s_part[w][0]; B += s_part[w][1]; T += s_part[w][2];
      C += s_part[w][3]; E += s_part[w][4]; F += s_part[w][5];
      if (s_part[w][6] > bv || (s_part[w][6] == bv && s_idx[w] < bi)) {
        bv = s_part[w][6]; bi = s_idx[w];
      }
    }
    const int nxt = bi;  // always < DTOK  ->  u_seq == 1 identically
    float lpn = lpr[nxt], lnn = lnr[nxt], lqn = lqr[nxt];
    float pair = (st == 0) ? (lnn + lqn) : (st == 1) ? (lpn + lqn) : (lpn + lnn);
    RowOut o;
    o.A = A; o.B = B; o.T = T; o.C = C; o.E = E; o.Fs = F;
    o.Es   = s_eos[0] + s_eos[1] + s_eos[2];
    o.sum3 = lpn + lnn + lqn;
    o.sm   = __expf(pair);
    out[row] = o;
  }
}

__global__ __launch_bounds__(BSZ)
void finalize_kernel(const RowOut* __restrict__ rows,
                     float* __restrict__ out, KLConsts kc) {
  const int b = threadIdx.x;
  RowOut r[MAXD];
#pragma unroll
  for (int d = 0; d < MAXD; ++d) r[d] = rows[d * BSZ + b];

  __shared__ float s_kl[BSZ / 32][MAXD];
  __shared__ float s_klm[MAXD];
  __shared__ float s_R[BSZ / 32];

  // KL batch means: kl(d) = 3*sum t log t  -  mean_b [c1*Fs + a*Es]
  float kp[MAXD];
#pragma unroll
  for (int d = 0; d < MAXD; ++d) kp[d] = kc.c1[d] * r[d].Fs + kc.a[d] * r[d].Es;
#pragma unroll
  for (int d = 0; d < MAXD; ++d) kp[d] = wave_sum(kp[d]);
  const int wid = b >> 5, lane = b & 31;
  if (lane == 0) {
#pragma unroll
    for (int d = 0; d < MAXD; ++d) s_kl[wid][d] = kp[d];
  }
  __syncthreads();
  if (b == 0) {
#pragma unroll
    for (int d = 0; d < MAXD; ++d) {
      float s = 0.f;
      for (int w = 0; w < BSZ / 32; ++w) s += s_kl[w][d];
      s_klm[d] = kc.t3[d] - s / (float)BSZ;
    }
  }
  __syncthreads();

  // base(d) = middle + last - ALPHA*kl ; middle = S*A + B ; last = T*C - E
  float S = 0.f, base[MAXD];
#pragma unroll
  for (int d = 0; d < MAXD; ++d) {
    base[d] = S * r[d].A + r[d].B + (r[d].T * r[d].C - r[d].E)
              - ALPHA_F * s_klm[d];
    S += r[d].sum3;
  }
  float R = base[MAXD - 1];                 // u_seq == 1 at every depth
#pragma unroll
  for (int d = MAXD - 2; d >= 0; --d) R = R * r[d].sm + base[d];

  R = wave_sum(R);
  if (lane == 0) s_R[wid] = R;
  __syncthreads();
  if (b == 0) {
    float s = 0.f;
    for (int w = 0; w < BSZ / 32; ++w) s += s_R[w];
    out[0] = -s / (float)BSZ;
  }
}

extern "C" void kernel_launch(void* const* d_in, const int* in_sizes, int n_in,
                              void* d_out, int out_size, void* d_ws, size_t ws_size,
                              hipStream_t stream) {
  (void)in_sizes; (void)n_in; (void)out_size; (void)ws_size;
  const float* lp = (const float*)d_in[0];
  const float* ln = (const float*)d_in[1];
  const float* lq = (const float*)d_in[2];
  const float* gm = (const float*)d_in[3];
  const int*   st = (const int*)d_in[4];
  RowOut* rows = (RowOut*)d_ws;   // 2560 * 36 B ~= 92 KB scratch

  // Depth-dependent KL-target constants (deterministic host compute).
  KLConsts kc;
  for (int k = 0; k < MAXD; ++k) {
    int j = MAXD - k;                     // depth k uses sum_{i=0..j} D^i
    double denom = 0.0, p = 1.0;
    for (int i = 0; i <= j; ++i) { denom += p; p *= (double)DTOK; }
    double a    = 1.0 / denom;
    double tnon = (1.0 - a) / (double)DTOK;
    double tlt  = (double)DTOK * tnon * log(tnon) + a * log(a);
    kc.c1[k] = (float)tnon;
    kc.a[k]  = (float)a;
    kc.t3[k] = (float)(3.0 * tlt);
  }

  row_scan_kernel<<<MAXD * BSZ, BLK, 0, stream>>>(lp, ln, lq, gm, st, rows);
  finalize_kernel<<<1, BSZ, 0, stream>>>(rows, (float*)d_out, kc);
}